// TimeHomogeneousTransition_18245021073879
// MI455X (gfx1250) — compile-verified
//
#include <hip/hip_runtime.h>
#include <hip/hip_bf16.h>
#include <math.h>

// P[b] = expm(Q * t[b]) via scaling-and-squaring + order-10 Taylor (Horner),
// all matmuls done with V_WMMA_F32_16X16X4_F32 on 64x64 zero-padded tiles.
// One workgroup (512 thr = 16 waves) per branch length; each wave owns one
// 16x16 tile of the 64x64 product.
//
// v2 changes vs v1:
//  - Horner loop fully unrolled -> 1/k becomes literal constants (kills the
//    v_div_scale/v_rcp/v_div_fmas sequence that polluted the inner loop).
//  - tile_matmul now preloads all 16 B fragments before issuing WMMAs (lets
//    the scheduler use partial s_wait_dscnt) and runs TWO independent WMMA
//    accumulator chains (K split in half) to break the serial D->C dependency.

#define NS      61          // codon states
#define NP      64          // padded dimension
#define STRIDE  66          // LDS row stride in floats (bank de-conflict)
#define N_TAYLOR 10

typedef __attribute__((ext_vector_type(2))) float v2f;
typedef __attribute__((ext_vector_type(8))) float v8f;

// Load A fragments for tile-row ti from an LDS matrix (row-major, STRIDE).
// ISA layout, 32-bit A 16x4: lanes 0-15 -> K={0,1}, lanes 16-31 -> K={2,3},
// M = lane % 16, VGPR v -> K = 2*h + v.
__device__ __forceinline__ void load_afrag(v2f afrag[16], const float* Abuf,
                                           int ti, int lane) {
  const int r = lane & 15;
  const int h = lane >> 4;
  #pragma unroll
  for (int kt = 0; kt < 4; ++kt) {
    #pragma unroll
    for (int ks = 0; ks < 4; ++ks) {
      const int k = kt * 16 + ks * 4 + 2 * h;      // even -> 8B aligned pair
      const float* p = &Abuf[(ti * 16 + r) * STRIDE + k];
      v2f a;
      a.x = p[0];
      a.y = p[1];
      afrag[kt * 4 + ks] = a;
    }
  }
}

// acc = A(ti,:) * B(:,tj) over K=64 using 16 WMMA f32 16x16x4 steps.
// B layout (4x16): lanes 0-15 -> K rows {0,1}, lanes 16-31 -> {2,3}, N = lane%16.
// All B fragments loaded first; two independent accumulator chains.
__device__ __forceinline__ v8f tile_matmul(const v2f afrag[16],
                                           const float* Bbuf,
                                           int tj, int lane) {
  const int c = lane & 15;
  const int h = lane >> 4;

  v2f bfrag[16];
  #pragma unroll
  for (int kt = 0; kt < 4; ++kt) {
    #pragma unroll
    for (int ks = 0; ks < 4; ++ks) {
      const int k = kt * 16 + ks * 4 + 2 * h;
      v2f bf;
      bf.x = Bbuf[(k + 0) * STRIDE + tj * 16 + c];
      bf.y = Bbuf[(k + 1) * STRIDE + tj * 16 + c];
      bfrag[kt * 4 + ks] = bf;
    }
  }

  v8f acc0 = {};
  v8f acc1 = {};
  #pragma unroll
  for (int i = 0; i < 8; ++i) {
    acc0 = __builtin_amdgcn_wmma_f32_16x16x4_f32(
        false, afrag[i],     false, bfrag[i],     (short)0, acc0, false, false);
    acc1 = __builtin_amdgcn_wmma_f32_16x16x4_f32(
        false, afrag[8 + i], false, bfrag[8 + i], (short)0, acc1, false, false);
  }

  v8f acc;
  #pragma unroll
  for (int v = 0; v < 8; ++v) acc[v] = acc0[v] + acc1[v];
  return acc;
}

// D layout (16x16 f32): VGPR v -> row ti*16 + v + 8*h, col tj*16 + lane%16.
__device__ __forceinline__ void store_tile(float* Dbuf, v8f d, int ti, int tj,
                                           int lane, float mul, bool addI) {
  const int c = lane & 15;
  const int h = lane >> 4;
  #pragma unroll
  for (int v = 0; v < 8; ++v) {
    const int row = ti * 16 + v + 8 * h;
    const int col = tj * 16 + c;
    float val = d[v] * mul;
    if (addI && row == col) val += 1.0f;
    Dbuf[row * STRIDE + col] = val;
  }
}

__global__ __launch_bounds__(512) void
TimeHomogeneousTransition_expm_wmma_kernel(const float* __restrict__ Q,
                                           const float* __restrict__ T,
                                           float* __restrict__ out) {
  __shared__ float sA[NP * STRIDE];   // scaled generator X = Q*t / 2^s
  __shared__ float sP0[NP * STRIDE];  // ping
  __shared__ float sP1[NP * STRIDE];  // pong
  __shared__ float srow[NP];          // row sums for inf-norm

  const int tid  = threadIdx.x;
  const int lane = tid & 31;
  const int wave = tid >> 5;
  const int ti   = wave >> 2;
  const int tj   = wave & 3;
  const int b    = blockIdx.x;
  const float tb = T[b];

  // ---- 1. X = Q * t, zero-padded to 64x64 ----
  for (int idx = tid; idx < NP * NP; idx += 512) {
    const int r = idx >> 6;
    const int c = idx & 63;
    float v = 0.0f;
    if (r < NS && c < NS) v = Q[r * NS + c] * tb;
    sA[r * STRIDE + c] = v;
  }
  __syncthreads();

  // ---- 2. inf-norm -> squaring count s, scale = 2^-s so ||A|| <= 0.5 ----
  if (tid < NP) {
    float rs = 0.0f;
    for (int c = 0; c < NP; ++c) rs += fabsf(sA[tid * STRIDE + c]);
    srow[tid] = rs;
  }
  __syncthreads();
  float nrm = 0.0f;
  for (int r = 0; r < NP; ++r) nrm = fmaxf(nrm, srow[r]);
  int sqn = 0;
  if (nrm > 0.5f) {
    sqn = (int)ceilf(log2f(nrm * 2.0f));
    if (sqn > 30) sqn = 30;
  }
  const float scale = exp2f((float)(-sqn));

  for (int idx = tid; idx < NP * NP; idx += 512) {
    const int r = idx >> 6;
    const int c = idx & 63;
    sA[r * STRIDE + c] *= scale;
  }
  __syncthreads();

  // ---- 3. Horner init: P = I + A/N ----
  const float invN = 1.0f / (float)N_TAYLOR;
  for (int idx = tid; idx < NP * NP; idx += 512) {
    const int r = idx >> 6;
    const int c = idx & 63;
    float v = sA[r * STRIDE + c] * invN;
    if (r == c) v += 1.0f;
    sP0[r * STRIDE + c] = v;
  }
  __syncthreads();

  float* Pin  = sP0;
  float* Pout = sP1;

  // A is constant through the Horner recurrence: preload its fragments once.
  v2f afrag[16];
  load_afrag(afrag, sA, ti, lane);

  // ---- 4. Horner: P <- I + (A @ P) / k, k = N-1 .. 1 ----
  // Fully unrolled: 1/k are compile-time constants (no f32 division in loop).
  #pragma unroll
  for (int k = N_TAYLOR - 1; k >= 1; --k) {
    v8f acc = tile_matmul(afrag, Pin, tj, lane);
    store_tile(Pout, acc, ti, tj, lane, 1.0f / (float)k, /*addI=*/true);
    __syncthreads();
    float* t2 = Pin; Pin = Pout; Pout = t2;
  }

  // ---- 5. Squarings: P <- P @ P, s times ----
  for (int sq = 0; sq < sqn; ++sq) {
    load_afrag(afrag, Pin, ti, lane);
    v8f acc = tile_matmul(afrag, Pin, tj, lane);
    store_tile(Pout, acc, ti, tj, lane, 1.0f, /*addI=*/false);
    __syncthreads();
    float* t2 = Pin; Pin = Pout; Pout = t2;
  }

  // ---- 6. log(P) with floor, write 61x61 block ----
  {
    const int c = lane & 15;
    const int h = lane >> 4;
    #pragma unroll
    for (int v = 0; v < 8; ++v) {
      const int row = ti * 16 + v + 8 * h;
      const int col = tj * 16 + c;
      if (row < NS && col < NS) {
        const float p = Pin[row * STRIDE + col];
        out[(size_t)b * (NS * NS) + row * NS + col] =
            (p > 0.0f) ? logf(p) : -1e18f;
      }
    }
  }
}

extern "C" void kernel_launch(void* const* d_in, const int* in_sizes, int n_in,
                              void* d_out, int out_size, void* d_ws, size_t ws_size,
                              hipStream_t stream) {
  (void)n_in; (void)out_size; (void)d_ws; (void)ws_size;
  const float* Q = (const float*)d_in[0];   // [61*61] f32
  const float* T = (const float*)d_in[1];   // [8192]  f32
  float* out = (float*)d_out;               // [8192*61*61] f32
  const int n_times = in_sizes[1];

  hipLaunchKernelGGL(TimeHomogeneousTransition_expm_wmma_kernel,
                     dim3(n_times), dim3(512), 0, stream, Q, T, out);
}